// DynamicGAT_17325898072670
// MI455X (gfx1250) — compile-verified
//
#include <hip/hip_runtime.h>
#include <hip/hip_bf16.h>
#include <math.h>

// ---------------------------------------------------------------------------
// GATv2 (2 layers) + global mean pool + FC for MI455X (gfx1250, wave32).
// Dense node-feature GEMMs: V_WMMA_F32_16X16X4_F32, one 16xNc strip per wave,
// B staged in LDS, K-blocked loads (clause -> one wait -> WMMA burst).
// Edge softmax/aggregation: atomic-based scatter kernels.
// ---------------------------------------------------------------------------

typedef __attribute__((ext_vector_type(2))) float v2f;
typedef __attribute__((ext_vector_type(8))) float v8f;

#define NEG_SLOPE 0.2f

// -------- float <-> order-preserving unsigned (for atomicMax on f32) --------
__device__ __forceinline__ unsigned ford(float f) {
  unsigned u = __float_as_uint(f);
  return (u & 0x80000000u) ? ~u : (u | 0x80000000u);
}
__device__ __forceinline__ float unford(unsigned u) {
  unsigned v = (u & 0x80000000u) ? (u ^ 0x80000000u) : ~u;
  return __uint_as_float(v);
}

__device__ __forceinline__ void edge_ends(const int* __restrict__ ei, int E,
                                          int e, int& s, int& d) {
  if (e < E) { s = ei[e]; d = ei[E + e]; }
  else       { s = e - E; d = s; }
}

// ---------------------------------------------------------------------------
// GEMM: C[M x Nc] = A[M x K] * B[K x Nc], row-major. Nc = TN*16.
// One wave computes a 16 x Nc strip (TN accumulators), so each A row is
// loaded exactly once. B lives in LDS (<= 32KB), loaded once per block.
// ---------------------------------------------------------------------------
template <int TN>
__global__ __launch_bounds__(256) void gemm_wmma(
    const float* __restrict__ A, const float* __restrict__ B,
    float* __restrict__ C, int M, int K) {
  const int Nc = TN * 16;
  __shared__ float sB[8192];  // up to 128 x 64 f32

  // Cooperative B stage (all threads, before any wave-level early-out).
  {
    int total = K * Nc;
    for (int i = threadIdx.x; i < total; i += 256) sB[i] = B[i];
  }
  __syncthreads();

  int wid  = blockIdx.x * (blockDim.x >> 5) + (threadIdx.x >> 5);
  int lane = threadIdx.x & 31;
  if (wid * 16 >= M) return;

  int half = lane >> 4;         // 0: lanes 0-15, 1: lanes 16-31
  int idx  = lane & 15;
  int rowA = wid * 16 + idx;
  if (rowA >= M) rowA = M - 1;  // clamp: duplicate compute, never stored
  const float* Arow = A + (size_t)rowA * K;

  v8f acc[TN];
#pragma unroll
  for (int t = 0; t < TN; ++t) acc[t] = (v8f){};

  for (int kb = 0; kb < K; kb += 16) {
    v2f af[4];
    v2f bf[4][TN];
#pragma unroll
    for (int ks = 0; ks < 4; ++ks) {
      int k0 = kb + ks * 4 + half * 2;          // this lane's K pair (even)
      af[ks] = *(const v2f*)(Arow + k0);        // single b64 load
      const float* B0 = sB + k0 * Nc + idx;
#pragma unroll
      for (int t = 0; t < TN; ++t) {
        bf[ks][t].x = B0[t * 16];
        bf[ks][t].y = B0[Nc + t * 16];
      }
    }
#pragma unroll
    for (int ks = 0; ks < 4; ++ks)
#pragma unroll
      for (int t = 0; t < TN; ++t)
        acc[t] = __builtin_amdgcn_wmma_f32_16x16x4_f32(
            false, af[ks], false, bf[ks][t], (short)0, acc[t], false, false);
  }

#pragma unroll
  for (int v = 0; v < 8; ++v) {
    int row = wid * 16 + v + half * 8;
    if (row < M) {
#pragma unroll
      for (int t = 0; t < TN; ++t)
        C[(size_t)row * Nc + t * 16 + idx] = acc[t][v];
    }
  }
}

// ---------------------------------------------------------------------------
// Self-loop attr = per-node mean of incoming edge_attr.
// ---------------------------------------------------------------------------
__global__ void loop_accum(const int* __restrict__ ei, const float* __restrict__ eattr,
                           float* __restrict__ lsum, float* __restrict__ lcnt, int E) {
  int e = blockIdx.x * blockDim.x + threadIdx.x;
  if (e >= E) return;
  int d = ei[E + e];
#pragma unroll
  for (int k = 0; k < 16; ++k)
    atomicAdd(&lsum[(size_t)d * 16 + k], eattr[(size_t)e * 16 + k]);
  atomicAdd(&lcnt[d], 1.0f);
}

__global__ void loop_div(float* __restrict__ lsum, const float* __restrict__ lcnt, int N) {
  int t = blockIdx.x * blockDim.x + threadIdx.x;
  if (t >= N * 16) return;
  lsum[t] /= fmaxf(lcnt[t >> 4], 1.0f);
}

// ---------------------------------------------------------------------------
// Pass A: per-edge attention score (wave per edge; 32 lanes = 32 channels/head)
// score[e,h] = att_h . leaky_relu(x_l[src] + x_r[dst] + ea @ w_e)
// ---------------------------------------------------------------------------
template <int H>
__global__ void edge_score(const int* __restrict__ ei, const float* __restrict__ eattr,
                           const float* __restrict__ lattr,
                           const float* __restrict__ xl, const float* __restrict__ xr,
                           const float* __restrict__ we, const float* __restrict__ att,
                           float* __restrict__ score, unsigned* __restrict__ nodemax,
                           int E, int N) {
  int wid  = (blockIdx.x * blockDim.x + threadIdx.x) >> 5;
  int lane = threadIdx.x & 31;
  int Etot = E + N;
  if (wid >= Etot) return;
  int s, d; edge_ends(ei, E, wid, s, d);
  const float* ea = (wid < E) ? (eattr + (size_t)wid * 16)
                              : (lattr + (size_t)(wid - E) * 16);
  const int HC = H * 32;

  float ps[H];
#pragma unroll
  for (int j = 0; j < H; ++j) {
    int ch = j * 32 + lane;
    float ef = 0.f;
#pragma unroll
    for (int k = 0; k < 16; ++k) ef += ea[k] * we[k * HC + ch];
    float m = xl[(size_t)s * HC + ch] + xr[(size_t)d * HC + ch] + ef;
    float lr = (m > 0.f) ? m : NEG_SLOPE * m;
    ps[j] = lr * att[ch];
  }
#pragma unroll
  for (int j = 0; j < H; ++j)
    for (int off = 16; off; off >>= 1) ps[j] += __shfl_xor(ps[j], off, 32);

  if (lane == 0) {
#pragma unroll
    for (int j = 0; j < H; ++j) {
      score[(size_t)wid * H + j] = ps[j];
      atomicMax(&nodemax[(size_t)d * H + j], ford(ps[j]));
    }
  }
}

// Pass B: ex = exp(score - max[dst]); denom[dst] += ex (thread per edge).
template <int H>
__global__ void edge_exp(const int* __restrict__ ei, float* __restrict__ score,
                         const unsigned* __restrict__ nodemax,
                         float* __restrict__ denom, int E, int N) {
  int e = blockIdx.x * blockDim.x + threadIdx.x;
  int Etot = E + N;
  if (e >= Etot) return;
  int s, d; edge_ends(ei, E, e, s, d);
#pragma unroll
  for (int j = 0; j < H; ++j) {
    float sc = score[(size_t)e * H + j];
    float mx = unford(nodemax[(size_t)d * H + j]);
    float ex = expf(sc - mx);
    score[(size_t)e * H + j] = ex;
    atomicAdd(&denom[(size_t)d * H + j], ex);
  }
}

// Pass C: out[dst] += alpha * x_l[src]  (wave per edge).
template <int H>
__global__ void edge_aggr(const int* __restrict__ ei, const float* __restrict__ ex,
                          const float* __restrict__ denom, const float* __restrict__ xl,
                          float* __restrict__ out, int E, int N) {
  int wid  = (blockIdx.x * blockDim.x + threadIdx.x) >> 5;
  int lane = threadIdx.x & 31;
  int Etot = E + N;
  if (wid >= Etot) return;
  int s, d; edge_ends(ei, E, wid, s, d);
  const int HC = H * 32;
#pragma unroll
  for (int j = 0; j < H; ++j) {
    int ch = j * 32 + lane;
    float alpha = ex[(size_t)wid * H + j] /
                  fmaxf(denom[(size_t)d * H + j], 1e-16f);
    atomicAdd(&out[(size_t)d * HC + ch], alpha * xl[(size_t)s * HC + ch]);
  }
}

__global__ void bias_relu(float* __restrict__ h, const float* __restrict__ b,
                          int total, int HC) {
  int t = blockIdx.x * blockDim.x + threadIdx.x;
  if (t >= total) return;
  h[t] = fmaxf(h[t] + b[t % HC], 0.f);
}

// Global mean pool (thread per node-channel).
__global__ void pool_sum(const float* __restrict__ h2, const int* __restrict__ batch,
                         float* __restrict__ psum, float* __restrict__ pcnt, int N) {
  int t = blockIdx.x * blockDim.x + threadIdx.x;
  if (t >= N * 32) return;
  int i = t >> 5, c = t & 31;
  int g = batch[i];
  atomicAdd(&psum[(size_t)g * 32 + c], h2[(size_t)i * 32 + c]);
  if (c == 0) atomicAdd(&pcnt[g], 1.0f);
}

// Final FC: out[256 x 32] = (psum / cnt) @ w_fc + b_fc.
__global__ void final_fc(const float* __restrict__ psum, const float* __restrict__ pcnt,
                         const float* __restrict__ wfc, const float* __restrict__ bfc,
                         float* __restrict__ out) {
  int t = blockIdx.x * blockDim.x + threadIdx.x;
  if (t >= 256 * 32) return;
  int g = t >> 5, o = t & 31;
  float inv = 1.0f / fmaxf(pcnt[g], 1.0f);
  float acc = bfc[o];
#pragma unroll
  for (int c = 0; c < 32; ++c)
    acc += psum[g * 32 + c] * inv * wfc[c * 32 + o];
  out[t] = acc;
}

// ---------------------------------------------------------------------------
static inline char* carve(char*& p, size_t bytes) {
  char* r = p;
  p += (bytes + 255) & ~size_t(255);
  return r;
}

extern "C" void kernel_launch(void* const* d_in, const int* in_sizes, int n_in,
                              void* d_out, int out_size, void* d_ws, size_t ws_size,
                              hipStream_t stream) {
  const float* x    = (const float*)d_in[0];
  const int*   ei   = (const int*)d_in[1];
  const float* ea   = (const float*)d_in[2];
  const int*   bat  = (const int*)d_in[3];
  const float* w_l1 = (const float*)d_in[4];
  const float* w_r1 = (const float*)d_in[5];
  const float* w_e1 = (const float*)d_in[6];
  const float* att1 = (const float*)d_in[7];
  const float* b1   = (const float*)d_in[8];
  const float* w_l2 = (const float*)d_in[9];
  const float* w_r2 = (const float*)d_in[10];
  const float* w_e2 = (const float*)d_in[11];
  const float* att2 = (const float*)d_in[12];
  const float* b2   = (const float*)d_in[13];
  const float* w_fc = (const float*)d_in[14];
  const float* b_fc = (const float*)d_in[15];

  const int N = in_sizes[0] / 128;
  const int E = in_sizes[1] / 2;
  const int Etot = E + N;

  char* p = (char*)d_ws;
  float*    xl1   = (float*)carve(p, (size_t)N * 64 * 4);
  float*    xr1   = (float*)carve(p, (size_t)N * 64 * 4);
  float*    lattr = (float*)carve(p, (size_t)N * 16 * 4);  // sums -> means in place
  float*    lcnt  = (float*)carve(p, (size_t)N * 4);
  float*    sc1   = (float*)carve(p, (size_t)Etot * 2 * 4); // scores -> exp in place
  unsigned* nm1   = (unsigned*)carve(p, (size_t)N * 2 * 4);
  float*    dn1   = (float*)carve(p, (size_t)N * 2 * 4);
  float*    h1    = (float*)carve(p, (size_t)N * 64 * 4);   // layer-1 out -> relu
  float*    xl2   = (float*)carve(p, (size_t)N * 32 * 4);
  float*    xr2   = (float*)carve(p, (size_t)N * 32 * 4);
  float*    sc2   = (float*)carve(p, (size_t)Etot * 4);
  unsigned* nm2   = (unsigned*)carve(p, (size_t)N * 4);
  float*    dn2   = (float*)carve(p, (size_t)N * 4);
  float*    h2    = (float*)carve(p, (size_t)N * 32 * 4);
  float*    psum  = (float*)carve(p, (size_t)256 * 32 * 4);
  float*    pcnt  = (float*)carve(p, (size_t)256 * 4);

  // Zero the atomic accumulators (every call: graph replays must match).
  hipMemsetAsync(lattr, 0, (size_t)N * 16 * 4, stream);
  hipMemsetAsync(lcnt,  0, (size_t)N * 4,      stream);
  hipMemsetAsync(nm1,   0, (size_t)N * 2 * 4,  stream);
  hipMemsetAsync(dn1,   0, (size_t)N * 2 * 4,  stream);
  hipMemsetAsync(h1,    0, (size_t)N * 64 * 4, stream);
  hipMemsetAsync(nm2,   0, (size_t)N * 4,      stream);
  hipMemsetAsync(dn2,   0, (size_t)N * 4,      stream);
  hipMemsetAsync(h2,    0, (size_t)N * 32 * 4, stream);
  hipMemsetAsync(psum,  0, (size_t)256 * 32 * 4, stream);
  hipMemsetAsync(pcnt,  0, (size_t)256 * 4,    stream);

  const int T = 256;
  auto cdiv = [](long long a, long long b) { return (int)((a + b - 1) / b); };

  // Self-loop mean edge attrs.
  loop_accum<<<cdiv(E, T), T, 0, stream>>>(ei, ea, lattr, lcnt, E);
  loop_div<<<cdiv((long long)N * 16, T), T, 0, stream>>>(lattr, lcnt, N);

  // Layer-1 node transforms: [N,128]x[128,64] via WMMA (16x64 strip per wave).
  {
    int waves = (N + 15) / 16;
    int grid = cdiv((long long)waves * 32, T);
    gemm_wmma<4><<<grid, T, 0, stream>>>(x, w_l1, xl1, N, 128);
    gemm_wmma<4><<<grid, T, 0, stream>>>(x, w_r1, xr1, N, 128);
  }

  // Layer-1 attention.
  edge_score<2><<<cdiv((long long)Etot * 32, T), T, 0, stream>>>(ei, ea, lattr, xl1, xr1, w_e1, att1, sc1, nm1, E, N);
  edge_exp<2><<<cdiv(Etot, T), T, 0, stream>>>(ei, sc1, nm1, dn1, E, N);
  edge_aggr<2><<<cdiv((long long)Etot * 32, T), T, 0, stream>>>(ei, sc1, dn1, xl1, h1, E, N);
  bias_relu<<<cdiv((long long)N * 64, T), T, 0, stream>>>(h1, b1, N * 64, 64);

  // Layer-2 node transforms: [N,64]x[64,32] via WMMA (16x32 strip per wave).
  {
    int waves = (N + 15) / 16;
    int grid = cdiv((long long)waves * 32, T);
    gemm_wmma<2><<<grid, T, 0, stream>>>(h1, w_l2, xl2, N, 64);
    gemm_wmma<2><<<grid, T, 0, stream>>>(h1, w_r2, xr2, N, 64);
  }

  // Layer-2 attention (heads=1; concat=False w/ 1 head == identity mean).
  edge_score<1><<<cdiv((long long)Etot * 32, T), T, 0, stream>>>(ei, ea, lattr, xl2, xr2, w_e2, att2, sc2, nm2, E, N);
  edge_exp<1><<<cdiv(Etot, T), T, 0, stream>>>(ei, sc2, nm2, dn2, E, N);
  edge_aggr<1><<<cdiv((long long)Etot * 32, T), T, 0, stream>>>(ei, sc2, dn2, xl2, h2, E, N);
  bias_relu<<<cdiv((long long)N * 32, T), T, 0, stream>>>(h2, b2, N * 32, 32);

  // Global mean pool + FC head.
  pool_sum<<<cdiv((long long)N * 32, T), T, 0, stream>>>(h2, bat, psum, pcnt, N);
  final_fc<<<cdiv(256 * 32, T), T, 0, stream>>>(psum, pcnt, w_fc, b_fc, (float*)d_out);
}